// FAVORAttention_87213605912649
// MI455X (gfx1250) — compile-verified
//
#include <hip/hip_runtime.h>
#include <math.h>

typedef float v2f  __attribute__((ext_vector_type(2)));
typedef float v8f  __attribute__((ext_vector_type(8)));
typedef unsigned int u32x4 __attribute__((ext_vector_type(4)));
typedef int   i32x4 __attribute__((ext_vector_type(4)));
typedef int   i32x8 __attribute__((ext_vector_type(8)));

#define BB   4
#define NN   4096
#define DD   512
#define HH   8
#define DKK  64
#define MM   128
#define FM   256          // 2*M feature dim
#define KVC  80           // kv cols: 64 value dims + 1 denominator + 15 pad

#define WTILE_K   64                  // k-chunk staged per TDM transfer
#define WTILE_LD  66                  // padded LDS row stride (dwords)
#define WTILE_FLOATS (64 * WTILE_LD)  // 4224 floats = 16896 B per buffer

// ---- fp32 WMMA wrapper: D = A(16x4) * B(4x16) + C ----
__device__ __forceinline__ v8f wmma_f32(v2f a, v2f b, v8f c) {
    return __builtin_amdgcn_wmma_f32_16x16x4_f32(
        false, a, false, b, (short)0, c, false, false);
}

__device__ __forceinline__ v8f zero8() {
    v8f z;
#pragma unroll
    for (int e = 0; e < 8; ++e) z[e] = 0.0f;
    return z;
}

// ---- TDM: DMA a 64x64 fp32 tile (row stride 512 floats) into LDS ----------
// Destination layout in LDS: 64 rows, padded stride 66 dwords (pad_interval
// code 5 = 64 dwords, pad_amount code 1 = 2 dwords) to avoid bank conflicts.
__device__ __forceinline__ void tdm_load_w_tile(const float* gsrc,
                                                unsigned int lds_byte_off,
                                                unsigned int tensor_rows) {
    unsigned long long ga = (unsigned long long)(uintptr_t)gsrc;
    // D# group 0: count=1 | lds_addr | global_addr | type=2 ("image")
    u32x4 g0 = {
        (unsigned int)__builtin_amdgcn_readfirstlane(1u),
        (unsigned int)__builtin_amdgcn_readfirstlane(lds_byte_off),
        (unsigned int)__builtin_amdgcn_readfirstlane((unsigned int)ga),
        (unsigned int)__builtin_amdgcn_readfirstlane(
            (((unsigned int)(ga >> 32)) & 0x1FFFFFFu) | (2u << 30)) };
    // D# group 1: data_size=2 (4B), pad_enable, pad_interval=5 (64 dw),
    // pad_amount=1 (2 dw); tensor_dim0=512, tensor_dim1=rows;
    // tile_dim0=64, tile_dim1=64; tensor_dim0_stride=512.
    unsigned int q0 = (2u << 16) | (1u << 20) | (5u << 22) | (1u << 25);
    unsigned int q1 = (512u & 0xFFFFu) << 16;                     // td0 lo
    unsigned int q2 = (512u >> 16) | ((tensor_rows & 0xFFFFu) << 16);
    unsigned int q3 = (tensor_rows >> 16) | (64u << 16);          // tile_dim0
    unsigned int q4 = 64u;                                        // tile_dim1
    unsigned int q5 = 512u;                                       // stride0
    i32x8 g1 = { __builtin_amdgcn_readfirstlane((int)q0),
                 __builtin_amdgcn_readfirstlane((int)q1),
                 __builtin_amdgcn_readfirstlane((int)q2),
                 __builtin_amdgcn_readfirstlane((int)q3),
                 __builtin_amdgcn_readfirstlane((int)q4),
                 __builtin_amdgcn_readfirstlane((int)q5), 0, 0 };
    i32x4 gz4 = { 0, 0, 0, 0 };      // groups 2/3 unused for 2-D tensors
    i32x8 gz8 = { 0, 0, 0, 0, 0, 0, 0, 0 };
    __builtin_amdgcn_tensor_load_to_lds(g0, g1, gz4, gz4, gz8, 0);
}

// ---------------------------------------------------------------------------
// Generic projection GEMM with TDM-staged weights:
//   C(rows x cols) = A(rows x 512) @ W^T + bias,  W is (cols, 512)
// Block = 8 waves * 16 rows = 128 rows, one 64-col tile per block.
// Wave 0 DMAs the next 64x64 W chunk while all waves compute from LDS.
// ---------------------------------------------------------------------------
template <bool QKV_SCATTER>
__device__ __forceinline__ void proj_gemm_body(
    const float* __restrict__ A, const float* __restrict__ W,
    const float* __restrict__ bias, int colTilesPerRow, unsigned int wRows,
    float* __restrict__ o0, float* __restrict__ o1, float* __restrict__ o2) {
    extern __shared__ float ldsW[];              // 2 * WTILE_FLOATS
    const int tid  = threadIdx.x;
    const int lane = tid & 31;
    const int wv   = tid >> 5;                   // 0..7
    const int rowBlk = blockIdx.x / colTilesPerRow;
    const int colBlk = blockIdx.x % colTilesPerRow;
    const int r0 = rowBlk * 128 + wv * 16;
    const int c0 = colBlk * 64;
    const int r = lane & 15, g = lane >> 4;
    const bool issuer = (wv == 0);
    const int NCH = DD / WTILE_K;                // 8 chunks

    v8f acc[4];
#pragma unroll
    for (int j = 0; j < 4; ++j) acc[j] = zero8();

    const float* arow = A + (size_t)(r0 + r) * DD;

    if (issuer) {
        tdm_load_w_tile(W + (size_t)c0 * DD, 0u, wRows);
        __builtin_amdgcn_s_wait_tensorcnt(0);
    }
    __syncthreads();

    for (int c = 0; c < NCH; ++c) {
        if (issuer && (c + 1) < NCH)             // overlap DMA with compute
            tdm_load_w_tile(W + (size_t)c0 * DD + (c + 1) * WTILE_K,
                            (unsigned int)(((c + 1) & 1) * WTILE_FLOATS * 4),
                            wRows);
        const float* ldsb = ldsW + (c & 1) * WTILE_FLOATS;
        const int kc = c * WTILE_K;
        v2f aF[16];                              // preload A fragments
#pragma unroll
        for (int s = 0; s < 16; ++s)
            aF[s] = *(const v2f*)(arow + kc + 4 * s + 2 * g);
#pragma unroll
        for (int s = 0; s < 16; ++s) {
            const int kloc = 4 * s + 2 * g;
#pragma unroll
            for (int j = 0; j < 4; ++j) {
                v2f bf = *(const v2f*)(ldsb + (j * 16 + r) * WTILE_LD + kloc);
                acc[j] = wmma_f32(aF[s], bf, acc[j]);
            }
        }
        if (issuer) __builtin_amdgcn_s_wait_tensorcnt(0);
        __syncthreads();
    }

#pragma unroll
    for (int j = 0; j < 4; ++j) {
#pragma unroll
        for (int e = 0; e < 8; ++e) {
            const int row = r0 + e + 8 * g;
            const int col = c0 + j * 16 + r;
            const float val = acc[j][e] + bias[col];
            if (QKV_SCATTER) {
                const int which = col / DD;      // 0=q 1=k 2=v
                const int d  = col % DD;
                const int h  = d >> 6, dk = d & 63;
                const int b  = row >> 12, n = row & (NN - 1);
                float* dst = (which == 0) ? o0 : ((which == 1) ? o1 : o2);
                dst[(((size_t)(b * HH + h)) * NN + n) * DKK + dk] = val;
            } else {
                o0[(size_t)row * DD + col] = val;
            }
        }
    }
}

__global__ void qkv_kernel(const float* __restrict__ x,
                           const float* __restrict__ W,
                           const float* __restrict__ bias,
                           float* __restrict__ q, float* __restrict__ k,
                           float* __restrict__ v) {
    proj_gemm_body<true>(x, W, bias, (3 * DD) / 64, 3 * DD, q, k, v);
}

__global__ void outproj_kernel(const float* __restrict__ attn,
                               const float* __restrict__ W,
                               const float* __restrict__ bias,
                               float* __restrict__ out) {
    proj_gemm_body<false>(attn, W, bias, DD / 64, DD, out, nullptr, nullptr);
}

// ---------------------------------------------------------------------------
// phi feature map: proj = (X/dk^0.25) @ omega[h]^T, 8 waves x 16x16 tiles,
// LDS exchange for row-max shift; feats = norm*[exp(p-s), exp(-p-s)]/16.
// ---------------------------------------------------------------------------
__global__ void phi_kernel(const float* __restrict__ X,      // (B,H,N,dk)
                           const float* __restrict__ omega,  // (H,M,dk)
                           const unsigned char* __restrict__ mask,
                           float* __restrict__ phi) {        // (B,H,N,2M)
    __shared__ float proj[16][MM + 4];
    __shared__ float rnorm[16];
    __shared__ float rshift[16];

    const int tid  = threadIdx.x;
    const int lane = tid & 31;
    const int widx = tid >> 5;
    const int bh   = blockIdx.x >> 8;
    const int tile = blockIdx.x & 255;
    const int b    = bh >> 3;
    const int h    = bh & (HH - 1);
    const int n0   = tile * 16;
    const int r = lane & 15, g = lane >> 4;
    const float scale = 0.3535533905932738f;     // 64^-0.25

    const float* xrow = X + ((size_t)bh * NN + n0 + r) * DKK;
    const float* om   = omega + (size_t)h * MM * DKK;
    const int m0 = widx * 16;

    v2f aF[16];                                  // preload scaled A frags
#pragma unroll
    for (int s = 0; s < 16; ++s) {
        v2f a = *(const v2f*)(xrow + 4 * s + 2 * g);
        a.x *= scale; a.y *= scale;
        aF[s] = a;
    }
    v8f acc = zero8();
    v2f bC = *(const v2f*)(om + (size_t)(m0 + r) * DKK + 2 * g);
#pragma unroll
    for (int s = 0; s < 16; ++s) {               // distance-1 B prefetch
        v2f bN = bC;
        if (s < 15)
            bN = *(const v2f*)(om + (size_t)(m0 + r) * DKK + 4 * (s + 1) + 2 * g);
        acc = wmma_f32(aF[s], bC, acc);
        bC = bN;
    }
#pragma unroll
    for (int e = 0; e < 8; ++e) proj[e + 8 * g][m0 + r] = acc[e];

    if (tid < 16) {
        const float* xr = X + ((size_t)bh * NN + n0 + tid) * DKK;
        float s2 = 0.0f;
        for (int d = 0; d < DKK; ++d) { float t = xr[d] * scale; s2 += t * t; }
        float nm = fmaxf(__expf(-0.5f * s2), 1e-6f);
        float vfac = 1.0f;
        if (mask) vfac = mask[(size_t)b * NN + n0 + tid] ? 0.0f : 1.0f;
        rnorm[tid] = nm * (1.0f / 16.0f) * vfac;
    }
    __syncthreads();
    if (tid < 16) {
        float s = 0.0f;
        for (int m = 0; m < MM; ++m) s = fmaxf(s, fabsf(proj[tid][m]));
        rshift[tid] = s;
    }
    __syncthreads();

    float* out = phi + ((size_t)bh * NN + n0) * FM;
    for (int idx = tid; idx < 16 * FM; idx += blockDim.x) {
        const int rr = idx >> 8;
        const int cc = idx & 255;
        const float p = (cc < MM) ? proj[rr][cc] : -proj[rr][cc - MM];
        out[(size_t)rr * FM + cc] = __expf(p - rshift[rr]) * rnorm[rr];
    }
}

// ---------------------------------------------------------------------------
// kv(2M x 80) = k_phi^T @ [V | ones | 0] over K=N=4096 (col 64 = k_sum).
// Distance-1 register pipeline decouples loads from WMMA.
// ---------------------------------------------------------------------------
__global__ void kv_kernel(const float* __restrict__ kphi, // (BH,N,2M)
                          const float* __restrict__ v,    // (BH,N,dk)
                          float* __restrict__ kv) {       // (BH,2M,KVC)
    const int lane = threadIdx.x & 31;
    const int wid  = blockIdx.x * (blockDim.x >> 5) + (threadIdx.x >> 5);
    const int bh   = wid / 80;
    const int rem  = wid % 80;
    const int m0 = (rem / 5) * 16;
    const int c0 = (rem % 5) * 16;
    const int r = lane & 15, g = lane >> 4;
    const int col = c0 + r;

    const float* kp = kphi + (size_t)bh * NN * FM;
    const float* vp = v    + (size_t)bh * NN * DKK;

    v8f acc = zero8();
    v2f aC, bC;
    {
        const int kk = 2 * g;
        aC.x = kp[(size_t)kk * FM + m0 + r];
        aC.y = kp[(size_t)(kk + 1) * FM + m0 + r];
        float bx = 0.0f, by = 0.0f;
        if (col < DKK) { bx = vp[(size_t)kk * DKK + col];
                         by = vp[(size_t)(kk + 1) * DKK + col]; }
        else if (col == DKK) { bx = 1.0f; by = 1.0f; }
        bC.x = bx; bC.y = by;
    }
    for (int n0 = 0; n0 < NN; n0 += 4) {
        v2f aN = aC, bN = bC;
        if (n0 + 4 < NN) {
            const int kk = n0 + 4 + 2 * g;
            aN.x = kp[(size_t)kk * FM + m0 + r];
            aN.y = kp[(size_t)(kk + 1) * FM + m0 + r];
            float bx = 0.0f, by = 0.0f;
            if (col < DKK) { bx = vp[(size_t)kk * DKK + col];
                             by = vp[(size_t)(kk + 1) * DKK + col]; }
            else if (col == DKK) { bx = 1.0f; by = 1.0f; }
            bN.x = bx; bN.y = by;
        }
        acc = wmma_f32(aC, bC, acc);
        aC = aN; bC = bN;
    }
    float* out = kv + ((size_t)bh * FM + m0) * KVC;
#pragma unroll
    for (int e = 0; e < 8; ++e) out[(size_t)(e + 8 * g) * KVC + col] = acc[e];
}

// ---------------------------------------------------------------------------
// num(16 x 80) = q_phi @ kv, divide by den (col 64), write merged-head attn.
// ---------------------------------------------------------------------------
__global__ void num_kernel(const float* __restrict__ qphi, // (BH,N,2M)
                           const float* __restrict__ kv,   // (BH,2M,KVC)
                           float* __restrict__ attn) {     // (B,N,D)
    __shared__ float sm[16][KVC + 4];
    const int tid  = threadIdx.x;
    const int lane = tid & 31;
    const int ct   = tid >> 5;
    const int bh   = blockIdx.x >> 8;
    const int tile = blockIdx.x & 255;
    const int b = bh >> 3, h = bh & 7;
    const int n0 = tile * 16;
    const int r = lane & 15, g = lane >> 4;
    const int c0 = ct * 16;

    const float* qrow = qphi + ((size_t)bh * NN + n0 + r) * FM;
    const float* kvp  = kv + (size_t)bh * FM * KVC;

    v8f acc = zero8();
    v2f aC = *(const v2f*)(qrow + 2 * g);
    v2f bC;
    bC.x = kvp[(size_t)(2 * g) * KVC + c0 + r];
    bC.y = kvp[(size_t)(2 * g + 1) * KVC + c0 + r];
    for (int k0 = 0; k0 < FM; k0 += 4) {
        v2f aN = aC, bN = bC;
        if (k0 + 4 < FM) {
            const int kk = k0 + 4 + 2 * g;
            aN = *(const v2f*)(qrow + kk);
            bN.x = kvp[(size_t)kk * KVC + c0 + r];
            bN.y = kvp[(size_t)(kk + 1) * KVC + c0 + r];
        }
        acc = wmma_f32(aC, bC, acc);
        aC = aN; bC = bN;
    }
#pragma unroll
    for (int e = 0; e < 8; ++e) sm[e + 8 * g][c0 + r] = acc[e];
    __syncthreads();

    for (int idx = tid; idx < 16 * DKK; idx += blockDim.x) {
        const int rr = idx >> 6, d = idx & 63;
        const float den = sm[rr][DKK] + 1e-6f;
        attn[((size_t)b * NN + n0 + rr) * DD + h * DKK + d] = sm[rr][d] / den;
    }
}

// ---------------------------------------------------------------------------
extern "C" void kernel_launch(void* const* d_in, const int* in_sizes, int n_in,
                              void* d_out, int out_size, void* d_ws, size_t ws_size,
                              hipStream_t stream) {
    (void)in_sizes; (void)n_in; (void)out_size; (void)ws_size;
    const float*         x     = (const float*)d_in[0];
    const unsigned char* mask  = (const unsigned char*)d_in[1];
    const float*         Wqkv  = (const float*)d_in[2];
    const float*         bqkv  = (const float*)d_in[3];
    const float*         Wout  = (const float*)d_in[4];
    const float*         bout  = (const float*)d_in[5];
    const float*         omega = (const float*)d_in[6];
    float*               out   = (float*)d_out;

    float* ws = (float*)d_ws;
    const size_t qkvsz = (size_t)BB * HH * NN * DKK;
    const size_t phisz = (size_t)BB * HH * NN * FM;
    float* q    = ws;
    float* k    = q + qkvsz;
    float* v    = k + qkvsz;
    float* qphi = v + qkvsz;
    float* kphi = qphi + phisz;
    float* kvb  = kphi + phisz;
    float* attn = q;                              // q dead after phi pass

    const size_t ldsBytes = 2 * WTILE_FLOATS * sizeof(float);  // 33792 B

    // 1) QKV projection: 128 row-blocks x 24 col-tiles
    qkv_kernel<<<128 * ((3 * DD) / 64), 256, ldsBytes, stream>>>(
        x, Wqkv, bqkv, q, k, v);
    // 2) phi features for q (no mask) and k (masked)
    phi_kernel<<<BB * HH * (NN / 16), 256, 0, stream>>>(q, omega, nullptr, qphi);
    phi_kernel<<<BB * HH * (NN / 16), 256, 0, stream>>>(k, omega, mask, kphi);
    // 3) kv + fused k_sum: 2560 waves
    kv_kernel<<<(BB * HH * 16 * 5) / 8, 256, 0, stream>>>(kphi, v, kvb);
    // 4) numerator / denominator, merged-head attn
    num_kernel<<<BB * HH * (NN / 16), 160, 0, stream>>>(qphi, kvb, attn);
    // 5) output projection: 128 row-blocks x 8 col-tiles
    outproj_kernel<<<128 * (DD / 64), 256, ldsBytes, stream>>>(
        attn, Wout, bout, out);
}